// Encoder_Mamba_14010183319591
// MI455X (gfx1250) — compile-verified
//
#include <hip/hip_runtime.h>
#include <cstdint>
#include <cstddef>

// ---------------------------------------------------------------------------
// VMamba encoder block for MI455X (gfx1250, wave32, WMMA f32 16x16x4).
//   GEMMs (in-proj, x-proj, out-proj, 1x1 pw-conv) -> v_wmma_f32_16x16x4_f32
//   Selective scan -> per-lane register recurrence + global_prefetch_b8
// Workspace use: ~120 MB (everything L2-resident on MI455X's 192MB L2).
// ---------------------------------------------------------------------------

#define BATCH 2
#define CCH   64
#define OCH   128
#define HH    96
#define WW    96
#define LL    (HH*WW)        // 9216
#define HDIM  32
#define DIN   64
#define NST   16
#define RNK   2
#define KDIR  4
#define BBX   (BATCH*2)      // batch*2 branches folded together
#define CDX   (RNK+2*NST)    // 34 = dt(2) + B(16) + C(16)
#define EPSV  1e-5f

typedef float v2f __attribute__((ext_vector_type(2)));
typedef float v8f __attribute__((ext_vector_type(8)));

__device__ __forceinline__ v8f wmma4(v2f a, v2f b, v8f c) {
  // D = A(16x4 f32) * B(4x16 f32) + C(16x16 f32)
  return __builtin_amdgcn_wmma_f32_16x16x4_f32(false, a, false, b, (short)0, c,
                                               false, false);
}
__device__ __forceinline__ float sigm(float x) { return 1.f / (1.f + __expf(-x)); }
__device__ __forceinline__ float siluf(float x) { return x * sigm(x); }
__device__ __forceinline__ float softplusf(float x) {
  return (x > 20.f) ? x : log1pf(__expf(x));
}
// square transpose index: l = h*W+w  ->  w*H+h  (H==W==96)
__device__ __forceinline__ int g1i(int l) { return (l % HH) * WW + (l / HH); }
__device__ __forceinline__ int dir_gather(int k, int l) {
  switch (k & 3) {
    case 0:  return l;
    case 1:  return g1i(l);
    case 2:  return LL - 1 - l;
    default: return g1i(LL - 1 - l);
  }
}

// ---------------- stage 1: depthwise 3x3 on input ---------------------------
__global__ void k_dw33(const float* __restrict__ x, const float* __restrict__ w,
                       const float* __restrict__ b, float* __restrict__ xd) {
  int t = blockIdx.x * blockDim.x + threadIdx.x;
  if (t >= BATCH * CCH * LL) return;
  int l = t % LL; int bc = t / LL; int c = bc % CCH;
  int h = l / WW, wi = l % WW;
  const float* xp = x + (size_t)bc * LL;
  const float* wp = w + c * 9;
  float acc = b[c];
#pragma unroll
  for (int i = 0; i < 3; i++) {
    int h2 = h + i - 1;
    if ((unsigned)h2 >= HH) continue;
#pragma unroll
    for (int j = 0; j < 3; j++) {
      int w2 = wi + j - 1;
      if ((unsigned)w2 >= WW) continue;
      acc += wp[i * 3 + j] * xp[h2 * WW + w2];
    }
  }
  xd[t] = acc;
}

// ---------------- stage 2: LayerNorm over HD=32 per pixel -------------------
__global__ void k_ln32(const float* __restrict__ xd, const float* __restrict__ lw,
                       const float* __restrict__ lb, float* __restrict__ hln) {
  int row = blockIdx.x * blockDim.x + threadIdx.x;
  if (row >= BBX * LL) return;
  int l = row % LL; int bb = row / LL;
  int b = bb >> 1, p = bb & 1;
  const float* src = xd + (size_t)(b * CCH + p * HDIM) * LL + l;
  float v[HDIM];
  float s = 0.f;
#pragma unroll
  for (int c = 0; c < HDIM; c++) { v[c] = src[(size_t)c * LL]; s += v[c]; }
  float m = s * (1.f / HDIM);
  float q = 0.f;
#pragma unroll
  for (int c = 0; c < HDIM; c++) { float d = v[c] - m; q += d * d; }
  float inv = rsqrtf(q * (1.f / HDIM) + EPSV);
  float* dst = hln + (size_t)row * HDIM;
#pragma unroll
  for (int c = 0; c < HDIM; c++) dst[c] = (v[c] - m) * inv * lw[c] + lb[c];
}

// ---------------- stage 3: in-proj GEMM (M=BBX*L, K=32, N=128) --------------
__global__ void k_gemm_in(const float* __restrict__ A, const float* __restrict__ Bw,
                          float* __restrict__ Cout) {
  int lane = threadIdx.x & 31, wv = threadIdx.x >> 5;
  int lid = lane & 15, half = lane >> 4;
  int m0 = blockIdx.x * 16;
  int n0 = (blockIdx.y * 4 + wv) * 16;
  v8f acc = {};
  for (int kk = 0; kk < HDIM; kk += 4) {
    int k0 = kk + half * 2;
    v2f a, b;
    const float* ap = A + (size_t)(m0 + lid) * HDIM + k0;
    a.x = ap[0]; a.y = ap[1];
    const float* bp = Bw + (size_t)k0 * 128 + n0 + lid;
    b.x = bp[0]; b.y = bp[128];
    acc = wmma4(a, b, acc);
  }
#pragma unroll
  for (int r = 0; r < 8; r++)
    Cout[(size_t)(m0 + r + half * 8) * 128 + n0 + lid] = acc[r];
}

// ---------------- stage 4: dw 3x3 conv + SiLU on inner stream ---------------
__global__ void k_vssconv(const float* __restrict__ xz, const float* __restrict__ w,
                          const float* __restrict__ b, float* __restrict__ xc) {
  int t = blockIdx.x * blockDim.x + threadIdx.x;
  if (t >= BBX * DIN * LL) return;
  int d = t % DIN;
  int rest = t / DIN;
  int l = rest % LL; int bb = rest / LL;
  int h = l / WW, wi = l % WW;
  const float* base = xz + (size_t)bb * LL * 128;   // xp = first 64 cols of xz
  const float* wp = w + d * 9;
  float acc = b[d];
#pragma unroll
  for (int i = 0; i < 3; i++) {
    int h2 = h + i - 1;
    if ((unsigned)h2 >= HH) continue;
#pragma unroll
    for (int j = 0; j < 3; j++) {
      int w2 = wi + j - 1;
      if ((unsigned)w2 >= WW) continue;
      acc += wp[i * 3 + j] * base[(size_t)(h2 * WW + w2) * 128 + d];
    }
  }
  xc[(size_t)(bb * DIN + d) * LL + l] = siluf(acc);
}

// ---------------- stage 5: x-proj GEMM with direction gather ----------------
// rows = (bb,k,l); A[row][d] = xc[bb,d,g_k(l)]; B = xproj_w[k] (34x64); N pad 48
__global__ void k_gemm_xproj(const float* __restrict__ xc, const float* __restrict__ xw,
                             float* __restrict__ xdbl) {
  int lane = threadIdx.x & 31, wv = threadIdx.x >> 5;  // block = 96 (3 waves)
  int lid = lane & 15, half = lane >> 4;
  int m0 = blockIdx.x * 16;
  int n0 = wv * 16;
  int m = m0 + lid;
  int bbk = m / LL; int l = m % LL;
  int bb = bbk >> 2, k = bbk & 3;
  int g = dir_gather(k, l);
  const float* Abase = xc + (size_t)bb * DIN * LL + g;
  const float* Bbase = xw + (size_t)k * CDX * DIN;
  int c = n0 + lid;
  bool cok = c < CDX;
  v8f acc = {};
  for (int kk = 0; kk < DIN; kk += 4) {
    int d0 = kk + half * 2;
    v2f a, bv;
    a.x = Abase[(size_t)d0 * LL];
    a.y = Abase[(size_t)(d0 + 1) * LL];
    if (cok) { bv.x = Bbase[c * DIN + d0]; bv.y = Bbase[c * DIN + d0 + 1]; }
    else     { bv.x = 0.f; bv.y = 0.f; }
    acc = wmma4(a, bv, acc);
  }
  if (cok) {
#pragma unroll
    for (int r = 0; r < 8; r++)
      xdbl[(size_t)(m0 + r + half * 8) * CDX + c] = acc[r];
  }
}

// ---------------- stage 6: selective scan (latency-bound) -------------------
// one lane per (bb,k,d) channel: 16 states in registers, 9216 sequential steps
__global__ void k_scan(const float* __restrict__ xdbl, const float* __restrict__ xc,
                       const float* __restrict__ dtw, const float* __restrict__ dtb,
                       const float* __restrict__ alog, const float* __restrict__ Ds,
                       float* __restrict__ ys) {
  int t = blockIdx.x * blockDim.x + threadIdx.x;  // 0..1023
  int d = t & 63; int k = (t >> 6) & 3; int bb = t >> 8;
  int kd = k * DIN + d;
  float Arow[NST], h[NST];
#pragma unroll
  for (int n = 0; n < NST; n++) { Arow[n] = -__expf(alog[kd * NST + n]); h[n] = 0.f; }
  float w0 = dtw[kd * 2 + 0], w1 = dtw[kd * 2 + 1];
  float db = dtb[kd], Dv = Ds[kd];
  const float* row = xdbl + (size_t)(bb * KDIR + k) * LL * CDX;
  const float* xcb = xc + (size_t)(bb * DIN + d) * LL;
  float* yb = ys + (size_t)(bb * KDIR + k) * LL * DIN + d;
  for (int l = 0; l < LL; ++l) {
    const float* xr = row + (size_t)l * CDX;
    __builtin_prefetch(xr + 16 * CDX, 0, 0);          // global_prefetch_b8
    float u = xcb[dir_gather(k, l)];
    float dt = softplusf(xr[0] * w0 + xr[1] * w1 + db);
    float du = dt * u;
    float acc = 0.f;
#pragma unroll
    for (int n = 0; n < NST; n++) {
      float dA = __expf(dt * Arow[n]);
      h[n] = dA * h[n] + du * xr[2 + n];
      acc = fmaf(h[n], xr[2 + NST + n], acc);
    }
    yb[(size_t)l * DIN] = acc + Dv * u;
  }
}

// ---------------- stage 7: merge 4 scan directions --------------------------
__global__ void k_merge(const float* __restrict__ ys, float* __restrict__ ym) {
  int t = blockIdx.x * blockDim.x + threadIdx.x;
  if (t >= BBX * LL * DIN) return;
  int d = t % DIN; int rest = t / DIN;
  int l = rest % LL; int bb = rest / LL;
  int lf = LL - 1 - l;
  int lt = g1i(l);
  int ltf = LL - 1 - lt;
  const float* base = ys + (size_t)bb * KDIR * LL * DIN + d;
  float y = base[(size_t)(0 * LL + l)  * DIN]
          + base[(size_t)(2 * LL + lf) * DIN]
          + base[(size_t)(1 * LL + lt) * DIN]
          + base[(size_t)(3 * LL + ltf)* DIN];
  ym[(size_t)(bb * LL + l) * DIN + d] = y;
}

// ---------------- stage 8: LN over DI + SiLU(z) gate (wave per pixel) -------
__global__ void k_lngate(const float* __restrict__ ym, const float* __restrict__ xz,
                         const float* __restrict__ ow, const float* __restrict__ ob,
                         float* __restrict__ yg) {
  int lane = threadIdx.x & 31;
  int wv = threadIdx.x >> 5;
  int row = blockIdx.x * 8 + wv;
  const float* src = ym + (size_t)row * DIN;
  float y0 = src[lane], y1 = src[lane + 32];
  float s = y0 + y1, q = y0 * y0 + y1 * y1;
#pragma unroll
  for (int off = 16; off >= 1; off >>= 1) {
    s += __shfl_xor(s, off, 32);
    q += __shfl_xor(q, off, 32);
  }
  float m = s * (1.f / DIN);
  float inv = rsqrtf(q * (1.f / DIN) - m * m + EPSV);
  const float* zr = xz + (size_t)row * 128 + 64;
  float* dst = yg + (size_t)row * DIN;
  float z0 = zr[lane], z1 = zr[lane + 32];
  dst[lane]      = ((y0 - m) * inv * ow[lane]      + ob[lane])      * siluf(z0);
  dst[lane + 32] = ((y1 - m) * inv * ow[lane + 32] + ob[lane + 32]) * siluf(z1);
}

// ---------------- stage 9: out-proj GEMM (M=BBX*L, K=64, N=32) --------------
__global__ void k_gemm_out(const float* __restrict__ A, const float* __restrict__ Bw,
                           float* __restrict__ Cout) {
  int lane = threadIdx.x & 31, wv = threadIdx.x >> 5;  // block = 64 (2 waves)
  int lid = lane & 15, half = lane >> 4;
  int m0 = blockIdx.x * 16;
  int n0 = wv * 16;
  v8f acc = {};
  for (int kk = 0; kk < DIN; kk += 4) {
    int k0 = kk + half * 2;
    v2f a, b;
    const float* ap = A + (size_t)(m0 + lid) * DIN + k0;
    a.x = ap[0]; a.y = ap[1];
    const float* bp = Bw + (size_t)k0 * HDIM + n0 + lid;
    b.x = bp[0]; b.y = bp[HDIM];
    acc = wmma4(a, b, acc);
  }
#pragma unroll
  for (int r = 0; r < 8; r++)
    Cout[(size_t)(m0 + r + half * 8) * HDIM + n0 + lid] = acc[r];
}

// ---------------- stage 10: branch residual combine -------------------------
__global__ void k_combine(const float* __restrict__ xd, const float* __restrict__ vout,
                          const float* __restrict__ scale, float* __restrict__ xmr) {
  int t = blockIdx.x * blockDim.x + threadIdx.x;
  if (t >= BATCH * CCH * LL) return;
  int l = t % LL; int bc = t / LL;
  int b = bc / CCH; int c = bc % CCH;
  int bb = b * 2 + (c >> 5); int cp = c & 31;
  float sc = scale[0] + 1.f;   // ms_scale*xp + (xp + y@out_w)
  xmr[t] = sc * xd[t] + vout[(size_t)(bb * LL + l) * HDIM + cp];
}

// ---------------- stage 11: instance norm + leaky relu ----------------------
__global__ void k_inorm(float* __restrict__ xm, const float* __restrict__ gw,
                        const float* __restrict__ gb) {
  __shared__ float ss[256], sq[256];
  int bc = blockIdx.x; int c = bc % CCH;
  float* p = xm + (size_t)bc * LL;
  float s = 0.f, q = 0.f;
  for (int i = threadIdx.x; i < LL; i += 256) { float v = p[i]; s += v; q += v * v; }
  ss[threadIdx.x] = s; sq[threadIdx.x] = q; __syncthreads();
  for (int off = 128; off; off >>= 1) {
    if ((int)threadIdx.x < off) {
      ss[threadIdx.x] += ss[threadIdx.x + off];
      sq[threadIdx.x] += sq[threadIdx.x + off];
    }
    __syncthreads();
  }
  float m = ss[0] * (1.f / LL);
  float inv = rsqrtf(sq[0] * (1.f / LL) - m * m + EPSV);
  float w = gw[c], b = gb[c];
  for (int i = threadIdx.x; i < LL; i += 256) {
    float v = (p[i] - m) * inv * w + b;
    p[i] = v > 0.f ? v : 0.01f * v;
  }
}

// ---------------- stage 12: global avg+max pool of x ------------------------
__global__ void k_pool(const float* __restrict__ x, float* __restrict__ pooled) {
  __shared__ float ss[256], sm[256];
  int bc = blockIdx.x; int b = bc / CCH; int c = bc % CCH;
  const float* p = x + (size_t)bc * LL;
  float s = 0.f, mx = -__builtin_inff();
  for (int i = threadIdx.x; i < LL; i += 256) { float v = p[i]; s += v; mx = fmaxf(mx, v); }
  ss[threadIdx.x] = s; sm[threadIdx.x] = mx; __syncthreads();
  for (int off = 128; off; off >>= 1) {
    if ((int)threadIdx.x < off) {
      ss[threadIdx.x] += ss[threadIdx.x + off];
      sm[threadIdx.x] = fmaxf(sm[threadIdx.x], sm[threadIdx.x + off]);
    }
    __syncthreads();
  }
  if (threadIdx.x == 0) {
    pooled[b * 2 * CCH + c]       = ss[0] * (1.f / LL);
    pooled[b * 2 * CCH + CCH + c] = sm[0];
  }
}

// ---------------- stage 13: sigmoid channel gate (3x3 conv center tap) ------
__global__ void k_gate(const float* __restrict__ pooled, const float* __restrict__ w,
                       const float* __restrict__ bias, float* __restrict__ gate) {
  int t = threadIdx.x;            // 128 threads
  int b = t / CCH, c = t % CCH;
  float acc = bias[c];
  for (int ic = 0; ic < 2 * CCH; ic++)
    acc += w[(size_t)(c * 2 * CCH + ic) * 9 + 4] * pooled[b * 2 * CCH + ic];
  gate[t] = sigm(acc);
}

// ---------------- stage 14: xs_ = xm + gate*x -------------------------------
__global__ void k_xs(const float* __restrict__ xm, const float* __restrict__ gate,
                     const float* __restrict__ x, float* __restrict__ xsb) {
  int t = blockIdx.x * blockDim.x + threadIdx.x;
  if (t >= BATCH * CCH * LL) return;
  int bc = t / LL;
  xsb[t] = xm[t] + gate[bc] * x[t];
}

// ---------------- stage 15: axial dw convs + BN + ReLU -> skip --------------
__global__ void k_axial(const float* __restrict__ xsb,
                        const float* __restrict__ whw, const float* __restrict__ whb,
                        const float* __restrict__ www, const float* __restrict__ wwb,
                        const float* __restrict__ bnw, const float* __restrict__ bnb,
                        const float* __restrict__ bnm, const float* __restrict__ bnv,
                        float* __restrict__ skip) {
  int t = blockIdx.x * blockDim.x + threadIdx.x;
  if (t >= BATCH * CCH * LL) return;
  int l = t % LL; int bc = t / LL; int c = bc % CCH;
  int h = l / WW, wi = l % WW;
  const float* p = xsb + (size_t)bc * LL;
  float vh = whb[c];
  if (h > 0)      vh += whw[c * 3 + 0] * p[l - WW];
  vh += whw[c * 3 + 1] * p[l];
  if (h < HH - 1) vh += whw[c * 3 + 2] * p[l + WW];
  float vw = wwb[c];
  if (wi > 0)      vw += www[c * 3 + 0] * p[l - 1];
  vw += www[c * 3 + 1] * p[l];
  if (wi < WW - 1) vw += www[c * 3 + 2] * p[l + 1];
  float v = p[l] + vh + vw;
  v = (v - bnm[c]) * rsqrtf(bnv[c] + EPSV) * bnw[c] + bnb[c];
  skip[t] = v > 0.f ? v : 0.f;
}

// ---------------- stage 16: 1x1 conv GEMM (M=B*L, K=64, N=128) --------------
__global__ void k_gemm_pw(const float* __restrict__ skip, const float* __restrict__ pw,
                          const float* __restrict__ pb, float* __restrict__ outp) {
  int lane = threadIdx.x & 31, wv = threadIdx.x >> 5;
  int lid = lane & 15, half = lane >> 4;
  int m0 = blockIdx.x * 16;
  int n0 = (blockIdx.y * 4 + wv) * 16;
  int m = m0 + lid; int b = m / LL; int l = m % LL;
  const float* Abase = skip + (size_t)b * CCH * LL + l;   // A[m][c]=skip[b,c,l]
  int oc = n0 + lid;
  v8f acc = {};
  for (int kk = 0; kk < CCH; kk += 4) {
    int c0 = kk + half * 2;
    v2f a, bv;
    a.x = Abase[(size_t)c0 * LL];
    a.y = Abase[(size_t)(c0 + 1) * LL];
    bv.x = pw[(size_t)oc * CCH + c0];
    bv.y = pw[(size_t)oc * CCH + c0 + 1];
    acc = wmma4(a, bv, acc);
  }
  float bias = pb[oc];
#pragma unroll
  for (int r = 0; r < 8; r++)
    outp[(size_t)(m0 + r + half * 8) * OCH + oc] = acc[r] + bias;
}

// ---------------- stage 17: 2x2 maxpool -> down -----------------------------
__global__ void k_maxpool(const float* __restrict__ outp, float* __restrict__ down) {
  int t = blockIdx.x * blockDim.x + threadIdx.x;
  if (t >= BATCH * OCH * 48 * 48) return;
  int ow = t % 48; int r = t / 48; int oh = r % 48; r /= 48;
  int oc = r % OCH; int b = r / OCH;
  int l0 = (2 * oh) * WW + 2 * ow;
  const float* base = outp + (size_t)b * LL * OCH + oc;
  float v = fmaxf(fmaxf(base[(size_t)l0 * OCH],        base[(size_t)(l0 + 1) * OCH]),
                  fmaxf(base[(size_t)(l0 + WW) * OCH], base[(size_t)(l0 + WW + 1) * OCH]));
  down[t] = v;
}

// ---------------------------------------------------------------------------
extern "C" void kernel_launch(void* const* d_in, const int* in_sizes, int n_in,
                              void* d_out, int out_size, void* d_ws, size_t ws_size,
                              hipStream_t stream) {
  (void)in_sizes; (void)n_in; (void)out_size; (void)ws_size;
  const float* x          = (const float*)d_in[0];
  const float* ms_dw33_w  = (const float*)d_in[1];
  const float* ms_dw33_b  = (const float*)d_in[2];
  const float* ms_in_w    = (const float*)d_in[3];
  const float* ms_in_b    = (const float*)d_in[4];
  const float* ms_scale   = (const float*)d_in[5];
  const float* vln_w      = (const float*)d_in[6];
  const float* vln_b      = (const float*)d_in[7];
  const float* vin_w      = (const float*)d_in[8];
  const float* vconv_w    = (const float*)d_in[9];
  const float* vconv_b    = (const float*)d_in[10];
  const float* vxproj_w   = (const float*)d_in[11];
  const float* vdtproj_w  = (const float*)d_in[12];
  const float* vdtproj_b  = (const float*)d_in[13];
  const float* vA_logs    = (const float*)d_in[14];
  const float* vDs        = (const float*)d_in[15];
  const float* von_w      = (const float*)d_in[16];
  const float* von_b      = (const float*)d_in[17];
  const float* vout_w     = (const float*)d_in[18];
  const float* ma_conv_w  = (const float*)d_in[19];
  const float* ma_conv_b  = (const float*)d_in[20];
  const float* adw_h_w    = (const float*)d_in[21];
  const float* adw_h_b    = (const float*)d_in[22];
  const float* adw_w_w    = (const float*)d_in[23];
  const float* adw_w_b    = (const float*)d_in[24];
  const float* pw_w       = (const float*)d_in[25];
  const float* pw_b       = (const float*)d_in[26];
  const float* bn_w       = (const float*)d_in[27];
  const float* bn_b       = (const float*)d_in[28];
  const float* bn_mean    = (const float*)d_in[29];
  const float* bn_var     = (const float*)d_in[30];

  float* ws = (float*)d_ws;
  size_t o = 0;
  float* xd   = ws + o; o += (size_t)BATCH * CCH * LL;        // 1.18M
  float* hln  = ws + o; o += (size_t)BBX * LL * HDIM;         // 1.18M
  float* xz   = ws + o; o += (size_t)BBX * LL * 128;          // 4.72M
  float* xc   = ws + o; o += (size_t)BBX * DIN * LL;          // 2.36M
  float* xdbl = ws + o; o += (size_t)BBX * KDIR * LL * CDX;   // 5.01M
  float* ys   = ws + o; o += (size_t)BBX * KDIR * LL * DIN;   // 9.44M
  float* ym   = ws + o; o += (size_t)BBX * LL * DIN;          // 2.36M
  float* yg   = ws + o; o += (size_t)BBX * LL * DIN;          // 2.36M
  float* vout = ws + o; o += (size_t)BBX * LL * HDIM;         // 1.18M
  float* pooled = ws + o; o += (size_t)BATCH * 2 * CCH;
  float* gateb  = ws + o; o += (size_t)BATCH * CCH;
  // safe aliases (lifetimes don't overlap):
  float* xmr  = hln;    // hln dead after gemm_in
  float* xsb  = xc;     // xc dead after scan
  float* outp = xz;     // xz dead after lngate

  float* down = (float*)d_out;
  float* skip = (float*)d_out + (size_t)BATCH * OCH * 48 * 48;

  const int NBCL = (BATCH * CCH * LL) / 256;       // 4608

  k_dw33   <<<NBCL, 256, 0, stream>>>(x, ms_dw33_w, ms_dw33_b, xd);
  k_ln32   <<<(BBX * LL) / 256, 256, 0, stream>>>(xd, vln_w, vln_b, hln);
  k_gemm_in<<<dim3((BBX * LL) / 16, 2), 128, 0, stream>>>(hln, vin_w, xz);
  k_vssconv<<<(BBX * DIN * LL) / 256, 256, 0, stream>>>(xz, vconv_w, vconv_b, xc);
  k_gemm_xproj<<<(BBX * KDIR * LL) / 16, 96, 0, stream>>>(xc, vxproj_w, xdbl);
  k_scan   <<<4, 256, 0, stream>>>(xdbl, xc, vdtproj_w, vdtproj_b, vA_logs, vDs, ys);
  k_merge  <<<(BBX * LL * DIN) / 256, 256, 0, stream>>>(ys, ym);
  k_lngate <<<(BBX * LL) / 8, 256, 0, stream>>>(ym, xz, von_w, von_b, yg);
  k_gemm_out<<<(BBX * LL) / 16, 64, 0, stream>>>(yg, vout_w, vout);
  k_combine<<<NBCL, 256, 0, stream>>>(xd, vout, ms_scale, xmr);
  k_inorm  <<<BATCH * CCH, 256, 0, stream>>>(xmr, ms_in_w, ms_in_b);
  k_pool   <<<BATCH * CCH, 256, 0, stream>>>(x, pooled);
  k_gate   <<<1, BATCH * CCH, 0, stream>>>(pooled, ma_conv_w, ma_conv_b, gateb);
  k_xs     <<<NBCL, 256, 0, stream>>>(xmr, gateb, x, xsb);
  k_axial  <<<NBCL, 256, 0, stream>>>(xsb, adw_h_w, adw_h_b, adw_w_w, adw_w_b,
                                      bn_w, bn_b, bn_mean, bn_var, skip);
  k_gemm_pw<<<dim3((BATCH * LL) / 16, 2), 128, 0, stream>>>(skip, pw_w, pw_b, outp);
  k_maxpool<<<(BATCH * OCH * 48 * 48) / 256, 256, 0, stream>>>(outp, down);
}